// WaveletTransform_46299747450906
// MI455X (gfx1250) — compile-verified
//
#include <hip/hip_runtime.h>

// Haar wavedec(max level) + waverec on (64,128,8192) fp32.
// Levels 1..4 (analysis) and 4..1 (synthesis) are done as 16x16 orthonormal
// matrix multiplies on v_wmma_f32_16x16x4_f32; levels 5..13 run on the 512
// approx coefficients in LDS with the reference's exact pairwise arithmetic.

typedef float v2f __attribute__((ext_vector_type(2)));
typedef float v8f __attribute__((ext_vector_type(8)));

#define INV_SQRT2F 0.70710678118654752440f
#define S3F        0.35355339059327376220f

// 4-level Haar analysis matrix H (16x16), orthonormal. Row 0 = approx row.
__device__ __forceinline__ float haar16(int r, int c) {
  if (r == 0) return 0.25f;
  if (r == 1) return (c < 8) ? 0.25f : -0.25f;
  if (r < 4) { int w = c - 8 * (r - 2);
    return (w < 0 || w >= 8) ? 0.0f : ((w < 4) ? S3F : -S3F); }
  if (r < 8) { int w = c - 4 * (r - 4);
    return (w < 0 || w >= 4) ? 0.0f : ((w < 2) ? 0.5f : -0.5f); }
  int w = c - 2 * (r - 8);
  return (w < 0 || w >= 2) ? 0.0f : ((w == 0) ? INV_SQRT2F : -INV_SQRT2F);
}

__global__ __launch_bounds__(256)
void WaveletTransform_46299747450906_kernel(const float* __restrict__ in,
                                            float* __restrict__ out) {
  __shared__ float xs[8192];    // signal / coefficient workspace (32 KB)
  __shared__ float amid[512];   // approx stream for levels 5..13
  __shared__ float dmid[512];   // detail storage for levels 5..13
  __shared__ float hmat[256];   // H (16x16)

  const int tid   = threadIdx.x;
  const int lane  = tid & 31;
  const int wave  = tid >> 5;          // 8 waves of 32 (wave32)
  const int laneM = lane & 15;
  const int hi    = (lane >> 4) & 1;   // half-wave select

  const size_t sig = blockIdx.x;       // one workgroup per (batch,channel) signal
  const float4* __restrict__ in4 = reinterpret_cast<const float4*>(in) + sig * 2048;
  float4* __restrict__ out4      = reinterpret_cast<float4*>(out) + sig * 2048;
  float4* xs4 = reinterpret_cast<float4*>(xs);

  // ---- global -> LDS, coalesced 16B per lane ----
  #pragma unroll
  for (int i = 0; i < 8; ++i) xs4[tid + 256 * i] = in4[tid + 256 * i];
  if (tid < 256) hmat[tid] = haar16(tid >> 4, tid & 15);
  __syncthreads();

  // ---- analysis levels 1..4: per tile (16 blocks x 16 samples) Y = H * X ----
  // Tile t covers samples [t*256, t*256+256); column j of X = block (t*16+j).
  #pragma unroll
  for (int i = 0; i < 4; ++i) {
    const int base = (wave * 4 + i) << 8;
    v8f acc = {};
    #pragma unroll
    for (int kk = 0; kk < 4; ++kk) {
      const int kA = 4 * kk + 2 * hi;   // this half-wave's K pair
      v2f a, b;
      a.x = hmat[laneM * 16 + kA];                // A[M][K] = H[M][k]
      a.y = hmat[laneM * 16 + kA + 1];
      b.x = xs[base + laneM * 16 + kA];           // B[K][N] = X[k][j]
      b.y = xs[base + laneM * 16 + kA + 1];
      acc = __builtin_amdgcn_wmma_f32_16x16x4_f32(false, a, false, b,
                                                  (short)0, acc, false, false);
    }
    // D: vgpr r -> row r (lanes 0-15) / row r+8 (lanes 16-31), col = laneM.
    // Store Y row-major within the tile (overwrites this wave's own X tile).
    #pragma unroll
    for (int r = 0; r < 8; ++r)
      xs[base + (r + 8 * hi) * 16 + laneM] = acc[r];
  }
  __syncthreads();

  // ---- gather approx stream: a[g] = Y_tile(g/16) row0 col (g%16) ----
  if (tid < 512) amid[tid] = xs[(tid >> 4) * 256 + (tid & 15)];
  __syncthreads();

  // ---- levels 5..13 analysis on 512 approx coefficients ----
  int off = 0;
  for (int n = 512; n > 1; n >>= 1) {
    const int m = n >> 1;
    const bool act = tid < m;
    float e = 0.f, o = 0.f;
    if (act) { e = amid[2 * tid]; o = amid[2 * tid + 1]; }
    __syncthreads();
    if (act) { amid[tid]       = (e + o) * INV_SQRT2F;
               dmid[off + tid] = (e - o) * INV_SQRT2F; }
    __syncthreads();
    off += m;                                    // ends at 511
  }
  // ---- levels 13..5 synthesis ----
  for (int m = 1; m <= 256; m <<= 1) {
    off -= m;
    const bool act = tid < m;
    float av = 0.f, dv = 0.f;
    if (act) { av = amid[tid]; dv = dmid[off + tid]; }
    __syncthreads();
    if (act) { amid[2 * tid]     = (av + dv) * INV_SQRT2F;
               amid[2 * tid + 1] = (av - dv) * INV_SQRT2F; }
    __syncthreads();
  }

  // ---- scatter approx back into row 0 of each Y tile ----
  if (tid < 512) xs[(tid >> 4) * 256 + (tid & 15)] = amid[tid];
  __syncthreads();

  // ---- synthesis levels 4..1: Z = H^T * Y ----
  #pragma unroll
  for (int i = 0; i < 4; ++i) {
    const int base = (wave * 4 + i) << 8;
    v8f acc = {};
    #pragma unroll
    for (int kk = 0; kk < 4; ++kk) {
      const int kB = 4 * kk + 2 * hi;
      v2f a, b;
      a.x = hmat[kB * 16 + laneM];                // A[M][k'] = H^T[M][k] = H[k][M]
      a.y = hmat[(kB + 1) * 16 + laneM];
      b.x = xs[base + kB * 16 + laneM];           // B[k'][j] = Y[k][j]
      b.y = xs[base + (kB + 1) * 16 + laneM];
      acc = __builtin_amdgcn_wmma_f32_16x16x4_f32(false, a, false, b,
                                                  (short)0, acc, false, false);
    }
    // Z[R][j] = sample (j*16 + R) of the tile; each lane owns 8 consecutive
    // floats -> two aligned float4 LDS stores.
    const int zb = base + laneM * 16 + 8 * hi;
    *reinterpret_cast<float4*>(&xs[zb])     = make_float4(acc[0], acc[1], acc[2], acc[3]);
    *reinterpret_cast<float4*>(&xs[zb + 4]) = make_float4(acc[4], acc[5], acc[6], acc[7]);
  }
  __syncthreads();

  // ---- LDS -> global, coalesced ----
  #pragma unroll
  for (int i = 0; i < 8; ++i) out4[tid + 256 * i] = xs4[tid + 256 * i];
}

extern "C" void kernel_launch(void* const* d_in, const int* in_sizes, int n_in,
                              void* d_out, int out_size, void* d_ws, size_t ws_size,
                              hipStream_t stream) {
  const float* x = (const float*)d_in[0];
  float* y = (float*)d_out;
  const int W = 8192;
  const int signals = in_sizes[0] / W;   // 64 * 128 = 8192 workgroups
  WaveletTransform_46299747450906_kernel<<<dim3(signals), dim3(256), 0, stream>>>(x, y);
}